// OptionRetriever_91044716741172
// MI455X (gfx1250) — compile-verified
//
#include <hip/hip_runtime.h>
#include <hip/hip_bf16.h>

typedef __attribute__((ext_vector_type(2))) float v2f;
typedef __attribute__((ext_vector_type(8))) float v8f;

#define HEADS 12
#define HH    64
#define HDIM  768
#define QLEN  64
#define DLEN  200
#define NDOCS 8

// Async global->LDS copy of 16 bytes (CDNA5 GLOBAL_LOAD_ASYNC_TO_LDS_B128,
// tracked with ASYNCcnt). lds_off = byte offset within workgroup LDS
// (= low 32 bits of generic shared pointer per ISA aperture layout).
__device__ __forceinline__ void async_cp_b128(unsigned int lds_off, const void* gp)
{
    unsigned long long ga = (unsigned long long)(uintptr_t)gp;
    asm volatile("global_load_async_to_lds_b128 %0, %1, off"
                 :: "v"(lds_off), "v"(ga) : "memory");
}
__device__ __forceinline__ void wait_async_le3()
{
    asm volatile("s_wait_asynccnt 0x3" ::: "memory");
}
__device__ __forceinline__ void wait_async_0()
{
    asm volatile("s_wait_asynccnt 0x0" ::: "memory");
}

// One K=16 block tile of WMMA work (4 sub-steps x 4 accumulators).
__device__ __forceinline__ void wmma_tile(const float* __restrict__ Asp,
                                          const float* __restrict__ Bsp,
                                          int wm, int wn, int lm, int half,
                                          v8f acc[2][2])
{
    #pragma unroll
    for (int kk = 0; kk < 16; kk += 4) {
        // A frag 16x4 (ISA 7.12.2): lane=M row; VGPR0 K={0|2}, VGPR1 K={1|3}.
        v2f a0 = *(const v2f*)&Asp[(wm * 32 + lm) * 16 + kk + 2 * half];
        v2f a1 = *(const v2f*)&Asp[(wm * 32 + 16 + lm) * 16 + kk + 2 * half];
        // B frag 4x16: lane=N col; VGPR0 K={0|2}, VGPR1 K={1|3}.
        v2f b0, b1;
        b0.x = Bsp[(kk + 2 * half)     * 128 + wn * 32 + lm];
        b0.y = Bsp[(kk + 2 * half + 1) * 128 + wn * 32 + lm];
        b1.x = Bsp[(kk + 2 * half)     * 128 + wn * 32 + 16 + lm];
        b1.y = Bsp[(kk + 2 * half + 1) * 128 + wn * 32 + 16 + lm];
        acc[0][0] = __builtin_amdgcn_wmma_f32_16x16x4_f32(false, a0, false, b0, (short)0, acc[0][0], false, false);
        acc[0][1] = __builtin_amdgcn_wmma_f32_16x16x4_f32(false, a0, false, b1, (short)0, acc[0][1], false, false);
        acc[1][0] = __builtin_amdgcn_wmma_f32_16x16x4_f32(false, a1, false, b0, (short)0, acc[1][0], false, false);
        acc[1][1] = __builtin_amdgcn_wmma_f32_16x16x4_f32(false, a1, false, b1, (short)0, acc[1][1], false, false);
    }
}

// ---------------------------------------------------------------------------
// C[M,N] = A[M,K] @ B[K,N] + bias[N]   (fp32, V_WMMA_F32_16X16X4_F32)
// Block: 256 threads = 8 waves (wave32). Block tile 64(M) x 128(N), BK=16.
// Wave grid 2x4; each wave computes a 32x32 tile = 4 x (16x16) accumulators.
// Double-buffered LDS fed by async copies; last tile peeled so the
// steady-state loop is branch-free: issue -> wait(<=3) -> barrier -> WMMA.
// M % 64 == 0, N % 128 == 0, K % 16 == 0 (and K >= 32) for all call sites.
// ---------------------------------------------------------------------------
__global__ __launch_bounds__(256)
void gemm_f32_wmma(const float* __restrict__ A, const float* __restrict__ B,
                   const float* __restrict__ bias, float* __restrict__ C,
                   int M, int N, int K)
{
    __shared__ __align__(16) float As[2][64 * 16];
    __shared__ __align__(16) float Bs[2][16 * 128];

    const int t    = threadIdx.x;
    const int lane = t & 31;
    const int wave = t >> 5;
    const int half = lane >> 4;   // 0: lanes 0-15, 1: lanes 16-31
    const int lm   = lane & 15;
    const int wm   = wave >> 2;   // 0..1  -> M sub-tile of 32
    const int wn   = wave & 3;    // 0..3  -> N sub-tile of 32
    const int m0   = blockIdx.y * 64;
    const int n0   = blockIdx.x * 128;

    v8f acc[2][2];
    #pragma unroll
    for (int i = 0; i < 2; ++i)
        #pragma unroll
        for (int j = 0; j < 2; ++j)
            acc[i][j] = (v8f){0.f, 0.f, 0.f, 0.f, 0.f, 0.f, 0.f, 0.f};

    const int ar  = t >> 2;           // A tile row this thread copies (0..63)
    const int ac  = (t & 3) << 2;     // A tile col (float4)
    const int br0 = t >> 5;           // B tile rows (two b128 per thread)
    const int bc0 = (t & 31) << 2;

    // Per-thread LDS destinations (byte offsets) for both buffers.
    unsigned int ldsA[2], ldsB0[2], ldsB1[2];
    #pragma unroll
    for (int p = 0; p < 2; ++p) {
        ldsA[p]  = (unsigned int)(uintptr_t)&As[p][ar * 16 + ac];
        ldsB0[p] = (unsigned int)(uintptr_t)&Bs[p][br0 * 128 + bc0];
        ldsB1[p] = (unsigned int)(uintptr_t)&Bs[p][(br0 + 8) * 128 + bc0];
    }
    // Per-thread global sources (advance by 16 floats in K per tile).
    const float* gA  = A + (size_t)(m0 + ar) * K + ac;
    const float* gB0 = B + (size_t)br0 * N + n0 + bc0;
    const float* gB1 = B + (size_t)(br0 + 8) * N + n0 + bc0;

    // Prologue: stream tile 0 into buffer 0.
    async_cp_b128(ldsA[0],  gA);
    async_cp_b128(ldsB0[0], gB0);
    async_cp_b128(ldsB1[0], gB1);

    // Steady state: compute tile p while tile p^1 streams in (branch-free).
    int p = 0;
    for (int k0 = 0; k0 + 16 < K; k0 += 16, p ^= 1) {
        async_cp_b128(ldsA[p ^ 1],  gA  + (k0 + 16));
        async_cp_b128(ldsB0[p ^ 1], gB0 + (size_t)(k0 + 16) * N);
        async_cp_b128(ldsB1[p ^ 1], gB1 + (size_t)(k0 + 16) * N);
        wait_async_le3();       // current tile's 3 copies complete (in-order)
        __syncthreads();        // all waves' copies for tile p now resident
        wmma_tile(As[p], Bs[p], wm, wn, lm, half, acc);
        __syncthreads();        // done reading tile p; safe to overwrite
    }
    // Peeled last tile.
    wait_async_0();
    __syncthreads();
    wmma_tile(As[p], Bs[p], wm, wn, lm, half, acc);

    // Epilogue: C 16x16 layout = VGPR r holds M=r (lanes 0-15) / M=r+8 (16-31).
    #pragma unroll
    for (int i = 0; i < 2; ++i)
        #pragma unroll
        for (int j = 0; j < 2; ++j) {
            const int col = n0 + wn * 32 + j * 16 + lm;
            const float bv = bias[col];
            #pragma unroll
            for (int r = 0; r < 8; ++r) {
                const int row = m0 + wm * 32 + i * 16 + r + 8 * half;
                C[(size_t)row * N + col] = acc[i][j][r] + bv;
            }
        }
}

// ---------------------------------------------------------------------------
// Attention stage 1: per (bod, head) block of 64 threads (thread = query row).
// S = q k^T / sqrt(200); softmax over l; h_pre = S v + skip.
// k/v read from kv buffer (L2-resident) as uniform broadcast float4 loads.
// ---------------------------------------------------------------------------
__global__ __launch_bounds__(64)
void attn1_kernel(const float* __restrict__ qbuf, const float* __restrict__ kvbuf,
                  const float* __restrict__ skipbuf, float* __restrict__ hpre)
{
    __shared__ float ssm[QLEN * 201];   // padded stride 201 -> conflict-free
    const int bod = blockIdx.x / HEADS;
    const int h   = blockIdx.x % HEADS;
    const int bo  = bod / NDOCS;
    const int u   = threadIdx.x;        // query row 0..63

    const float* qrow = qbuf + (size_t)(bo * QLEN + u) * HDIM + h * HH;
    float4 qr[16];
    #pragma unroll
    for (int x4 = 0; x4 < 16; ++x4) qr[x4] = *(const float4*)&qrow[x4 * 4];

    const float* kvb = kvbuf + (size_t)bod * DLEN * (2 * HDIM);
    const float scale = 0.07071067811865475f;   // 1/sqrt(doc_len)

    float* srow = &ssm[u * 201];
    for (int l = 0; l < DLEN; ++l) {
        const float* kr = kvb + (size_t)l * (2 * HDIM) + h * (2 * HH);
        float dot = 0.f;
        #pragma unroll
        for (int x4 = 0; x4 < 16; ++x4) {
            float4 kk = *(const float4*)&kr[x4 * 4];
            dot += qr[x4].x * kk.x + qr[x4].y * kk.y + qr[x4].z * kk.z + qr[x4].w * kk.w;
        }
        srow[l] = dot * scale;
    }
    // Row softmax (row is private to this thread: no barrier needed).
    float mx = -__builtin_inff();
    for (int l = 0; l < DLEN; ++l) mx = fmaxf(mx, srow[l]);
    float sum = 0.f;
    for (int l = 0; l < DLEN; ++l) { float e = __expf(srow[l] - mx); srow[l] = e; sum += e; }
    const float inv = 1.f / sum;

    float4 acc[16];
    #pragma unroll
    for (int x4 = 0; x4 < 16; ++x4) acc[x4] = make_float4(0.f, 0.f, 0.f, 0.f);
    for (int l = 0; l < DLEN; ++l) {
        const float w = srow[l] * inv;
        const float* vr = kvb + (size_t)l * (2 * HDIM) + h * (2 * HH) + HH;
        #pragma unroll
        for (int x4 = 0; x4 < 16; ++x4) {
            float4 vv = *(const float4*)&vr[x4 * 4];
            acc[x4].x += w * vv.x; acc[x4].y += w * vv.y;
            acc[x4].z += w * vv.z; acc[x4].w += w * vv.w;
        }
    }
    const float* sk  = skipbuf + (size_t)(bo * QLEN + u) * HDIM + h * HH;
    float*       out = hpre    + (size_t)(bod * QLEN + u) * HDIM + h * HH;
    #pragma unroll
    for (int x4 = 0; x4 < 16; ++x4) {
        out[x4 * 4 + 0] = acc[x4].x + sk[x4 * 4 + 0];
        out[x4 * 4 + 1] = acc[x4].y + sk[x4 * 4 + 1];
        out[x4 * 4 + 2] = acc[x4].z + sk[x4 * 4 + 2];
        out[x4 * 4 + 3] = acc[x4].w + sk[x4 * 4 + 3];
    }
}

// ---------------------------------------------------------------------------
// In-place LayerNorm over rows of 768; one wave (32 lanes) per row.
// ---------------------------------------------------------------------------
__global__ __launch_bounds__(256)
void ln_kernel(float* __restrict__ h, const float* __restrict__ gamma,
               const float* __restrict__ beta)
{
    const int lane = threadIdx.x & 31;
    const int row  = blockIdx.x * 8 + (threadIdx.x >> 5);
    float* base = h + (size_t)row * HDIM;
    float s = 0.f, s2 = 0.f;
    for (int c = lane; c < HDIM; c += 32) { float v = base[c]; s += v; s2 += v * v; }
    #pragma unroll
    for (int o = 16; o > 0; o >>= 1) {
        s  += __shfl_xor(s,  o, 32);
        s2 += __shfl_xor(s2, o, 32);
    }
    const float mu  = s * (1.f / HDIM);
    const float var = s2 * (1.f / HDIM) - mu * mu;
    const float inv = rsqrtf(var + 1e-5f);
    for (int c = lane; c < HDIM; c += 32)
        base[c] = (base[c] - mu) * inv * gamma[c] + beta[c];
}

// ---------------------------------------------------------------------------
// Attention stage 2 + score: one block per (b,o,d).
// score[bod] = sum_{h,v} softmax_v(q2_h . k2_hv / 8)[v] * (sum_x v2[h,v,x])
// ---------------------------------------------------------------------------
__global__ __launch_bounds__(256)
void attn2_kernel(const float* __restrict__ qkv, const int* __restrict__ mask,
                  float* __restrict__ out)
{
    __shared__ float q2sm[HEADS * HH];
    __shared__ float w2sm[HEADS * QLEN];
    __shared__ float red[256];
    const int bod = blockIdx.x;
    const int bo  = bod / NDOCS;
    const int t   = threadIdx.x;
    const float* qb = qkv + (size_t)bod * QLEN * (3 * HDIM);   // rows of 2304

    for (int idx = t; idx < HEADS * HH; idx += 256) {
        const int h = idx >> 6, x = idx & 63;
        q2sm[idx] = qb[h * 192 + x];                 // q2 at query position u=0
    }
    __syncthreads();

    for (int idx = t; idx < HEADS * QLEN; idx += 256) {
        const int h = idx >> 6, v = idx & 63;
        const float* k2 = qb + (size_t)v * (3 * HDIM) + h * 192 + 64;
        float dot = 0.f;
        #pragma unroll 8
        for (int x = 0; x < HH; ++x) dot += q2sm[h * 64 + x] * k2[x];
        float lg = dot * 0.125f;                     // 1/sqrt(q_len)
        if (mask[bo * QLEN + v] == 0) lg = -__builtin_inff();
        w2sm[idx] = lg;
    }
    __syncthreads();

    const int wv = t >> 5, lane = t & 31;
    for (int h = wv; h < HEADS; h += 8) {            // per-head softmax over 64
        float a = w2sm[h * 64 + lane], b = w2sm[h * 64 + 32 + lane];
        float mx = fmaxf(a, b);
        #pragma unroll
        for (int o = 16; o > 0; o >>= 1) mx = fmaxf(mx, __shfl_xor(mx, o, 32));
        float ea = __expf(a - mx), eb = __expf(b - mx);
        float sm = ea + eb;
        #pragma unroll
        for (int o = 16; o > 0; o >>= 1) sm += __shfl_xor(sm, o, 32);
        const float inv = 1.f / sm;
        w2sm[h * 64 + lane]      = ea * inv;
        w2sm[h * 64 + 32 + lane] = eb * inv;
    }
    __syncthreads();

    float part = 0.f;
    for (int idx = t; idx < HEADS * QLEN; idx += 256) {
        const int h = idx >> 6, v = idx & 63;
        const float* v2 = qb + (size_t)v * (3 * HDIM) + h * 192 + 128;
        float s = 0.f;
        #pragma unroll 8
        for (int x = 0; x < HH; ++x) s += v2[x];
        part += w2sm[idx] * s;
    }
    red[t] = part;
    __syncthreads();
    #pragma unroll
    for (int o = 128; o > 0; o >>= 1) {
        if (t < o) red[t] += red[t + o];
        __syncthreads();
    }
    if (t == 0) out[bod] = red[0];
}

// ---------------------------------------------------------------------------
extern "C" void kernel_launch(void* const* d_in, const int* in_sizes, int n_in,
                              void* d_out, int out_size, void* d_ws, size_t ws_size,
                              hipStream_t stream)
{
    const float* hd    = (const float*)d_in[0];
    const float* hq    = (const float*)d_in[1];
    const int*   mask  = (const int*)  d_in[2];
    const float* Wq    = (const float*)d_in[3];
    const float* bq    = (const float*)d_in[4];
    const float* Wkv   = (const float*)d_in[5];
    const float* bkv   = (const float*)d_in[6];
    const float* Wres  = (const float*)d_in[7];
    const float* bres  = (const float*)d_in[8];
    const float* gamma = (const float*)d_in[9];
    const float* beta  = (const float*)d_in[10];
    const float* Wproj = (const float*)d_in[11];
    const float* bproj = (const float*)d_in[12];
    float* out = (float*)d_out;

    float* ws     = (float*)d_ws;
    float* qbuf   = ws;                     //  512 x  768
    float* skipb  = qbuf  + 512 * 768;      //  512 x  768
    float* hbuf   = skipb + 512 * 768;      // 4096 x  768
    float* kvbuf  = hbuf  + 4096 * 768;     // 12800 x 1536
    float* qkvbuf = kvbuf;                  // alias: 4096 x 2304 (kv dead by then)

    // q = hq @ Wq + bq ; skip = hq @ Wres + bres
    gemm_f32_wmma<<<dim3(6, 8),    256, 0, stream>>>(hq, Wq,   bq,   qbuf,  512,   768,  768);
    gemm_f32_wmma<<<dim3(6, 8),    256, 0, stream>>>(hq, Wres, bres, skipb, 512,   768,  768);
    // kv = hd @ Wkv + bkv   (dominant GEMM: 12800 x 1536 x 768)
    gemm_f32_wmma<<<dim3(12, 200), 256, 0, stream>>>(hd, Wkv,  bkv,  kvbuf, 12800, 1536, 768);
    // per-(doc,head) attention + skip add
    attn1_kernel<<<64 * HEADS, 64, 0, stream>>>(qbuf, kvbuf, skipb, hbuf);
    // in-place layernorm over H=768
    ln_kernel<<<512, 256, 0, stream>>>(hbuf, gamma, beta);
    // qkv = h @ Wproj + bproj  (4096 x 2304 x 768)
    gemm_f32_wmma<<<dim3(18, 64),  256, 0, stream>>>(hbuf, Wproj, bproj, qkvbuf, 4096, 2304, 768);
    // stage-2 attention + final score reduction
    attn2_kernel<<<64, 256, 0, stream>>>(qkvbuf, mask, out);
}